// DownsamplingLayer_40355512713475
// MI455X (gfx1250) — compile-verified
//
#include <hip/hip_runtime.h>
#include <stdint.h>

// Problem constants (fixed by the reference setup_inputs()).
#define B_DIM 64
#define N_DIM 262144     // 2^18
#define M_DIM 16384      // 2^14

#define RED_BLOCKS  1024
#define RED_THREADS 256
#define TILE_ELEMS  4096                       // 16 KB tile staged via TDM
#define TOTAL_WL    ((size_t)B_DIM * N_DIM)    // 16,777,216
#define TILES_TOTAL (TOTAL_WL / TILE_ELEMS)    // 4096
#define TILES_PER_BLOCK (TILES_TOTAL / RED_BLOCKS) // 4

typedef unsigned int u32;
typedef __attribute__((ext_vector_type(4))) unsigned int u32x4;
typedef __attribute__((ext_vector_type(8))) int i32x8;
typedef __attribute__((ext_vector_type(4))) int i32x4;

#if __has_builtin(__builtin_amdgcn_tensor_load_to_lds)
#define HAVE_TDM 1
#else
#define HAVE_TDM 0
#endif

// Order-preserving float <-> u32 key so min/max combine is integer atomicMin/Max.
__device__ __forceinline__ u32 f2key(float f) {
    u32 u = __float_as_uint(f);
    return (u >> 31) ? ~u : (u | 0x80000000u);
}
__device__ __forceinline__ float key2f(u32 k) {
    u32 u = (k >> 31) ? (k ^ 0x80000000u) : ~k;
    return __uint_as_float(u);
}

#if HAVE_TDM
// Issue one TDM DMA: contiguous TILE_ELEMS f32 from `src` into LDS byte offset `ldsAddr`.
// D# layout per CDNA5 ISA §8: group0 = {count/type/addr}, group1 = {dims/strides}.
__device__ __forceinline__ void tdm_issue_tile(const float* src, u32 ldsAddr) {
    const uint64_t ga = (uint64_t)(size_t)src;
    u32x4 g0;
    g0[0] = 1u;                                            // count=1, user D#
    g0[1] = ldsAddr;                                       // lds_addr (bytes)
    g0[2] = (u32)ga;                                       // global_addr[31:0]
    g0[3] = (u32)((ga >> 32) & 0x1FFFFFFu) | (2u << 30);   // global_addr[56:32], type=2
    i32x8 g1;
    g1[0] = (int)(2u << 16);                 // data_size=2 (4 bytes), no flags
    g1[1] = (int)((u32)TILE_ELEMS << 16);    // tensor_dim0[15:0] in [31:16]
    g1[2] = (int)(1u << 16);                 // tensor_dim0[31:16]=0, tensor_dim1=1
    g1[3] = (int)((u32)TILE_ELEMS << 16);    // tensor_dim1 hi=0, tile_dim0=TILE_ELEMS
    g1[4] = 1;                               // tile_dim1=1, tile_dim2=0
    g1[5] = (int)TILE_ELEMS;                 // tensor_dim0_stride[31:0]
    g1[6] = 0;                               // stride hi / tensor_dim1_stride lo
    g1[7] = 0;                               // tensor_dim1_stride hi
    i32x4 g2 = {0, 0, 0, 0};
    i32x4 g3 = {0, 0, 0, 0};
#if __clang_major__ >= 23
    i32x8 g4 = {0, 0, 0, 0, 0, 0, 0, 0};
    __builtin_amdgcn_tensor_load_to_lds(g0, g1, g2, g3, g4, 0);
#else
    __builtin_amdgcn_tensor_load_to_lds(g0, g1, g2, g3, 0);
#endif
}
#endif

// ---------------------------------------------------------------------------
// Pass 1: global min/max of high_res_wavelength, TDM double-buffered staging.
// ---------------------------------------------------------------------------
__global__ __launch_bounds__(RED_THREADS)
void ds_minmax_kernel(const float* __restrict__ wl, u32* __restrict__ ws) {
    __shared__ __align__(16) float tile[2][TILE_ELEMS];
    __shared__ u32 smin[RED_THREADS];
    __shared__ u32 smax[RED_THREADS];

    const int tid = threadIdx.x;
    u32 lmin = 0xFFFFFFFFu;
    u32 lmax = 0x00000000u;

    const size_t tileBase = (size_t)blockIdx.x * TILES_PER_BLOCK;

#if HAVE_TDM
    const u32 ldsAddr0 = (u32)(size_t)(&tile[0][0]);  // LDS aperture: low 32 bits = offset
    const u32 ldsAddr1 = (u32)(size_t)(&tile[1][0]);

    // Prime the pipeline: wave holding lane 0 issues DMA for tile 0 (TDM ignores EXEC).
    if (tid == 0) {
        tdm_issue_tile(wl + tileBase * TILE_ELEMS, ldsAddr0);
    }

    for (size_t t = 0; t < TILES_PER_BLOCK; ++t) {
        if (tid == 0) {
            if (t + 1 < TILES_PER_BLOCK) {
                // Overlap: issue t+1, then wait only until tile t has landed
                // (TENSORcnt completes in-order within a wave).
                tdm_issue_tile(wl + (tileBase + t + 1) * TILE_ELEMS,
                               ((t + 1) & 1) ? ldsAddr1 : ldsAddr0);
#if __has_builtin(__builtin_amdgcn_s_wait_tensorcnt)
                __builtin_amdgcn_s_wait_tensorcnt(1);
#else
                asm volatile("s_wait_tensorcnt 1" ::: "memory");
#endif
            } else {
#if __has_builtin(__builtin_amdgcn_s_wait_tensorcnt)
                __builtin_amdgcn_s_wait_tensorcnt(0);
#else
                asm volatile("s_wait_tensorcnt 0" ::: "memory");
#endif
            }
        }
        __syncthreads();  // publish tile t to all 8 waves

        const float4* t4 = (const float4*)&tile[t & 1][0];
        #pragma unroll
        for (int i = 0; i < TILE_ELEMS / 4 / RED_THREADS; ++i) {  // 4 x ds_load_b128
            const float4 v = t4[tid + i * RED_THREADS];
            u32 k0 = f2key(v.x), k1 = f2key(v.y), k2 = f2key(v.z), k3 = f2key(v.w);
            u32 kmin = min(min(k0, k1), min(k2, k3));
            u32 kmax = max(max(k0, k1), max(k2, k3));
            lmin = (kmin < lmin) ? kmin : lmin;
            lmax = (kmax > lmax) ? kmax : lmax;
        }
        __syncthreads();  // tile t free for reuse before DMA t+2 lands
    }
#else
    // Fallback: direct vectorized streaming loads (no TDM builtin on this toolchain).
    for (size_t t = 0; t < TILES_PER_BLOCK; ++t) {
        const float4* src4 = (const float4*)(wl + (tileBase + t) * TILE_ELEMS);
        for (int i = tid; i < TILE_ELEMS / 4; i += RED_THREADS) {
            const float4 v = src4[i];
            u32 k0 = f2key(v.x), k1 = f2key(v.y), k2 = f2key(v.z), k3 = f2key(v.w);
            u32 kmin = min(min(k0, k1), min(k2, k3));
            u32 kmax = max(max(k0, k1), max(k2, k3));
            lmin = (kmin < lmin) ? kmin : lmin;
            lmax = (kmax > lmax) ? kmax : lmax;
        }
    }
#endif

    smin[tid] = lmin;
    smax[tid] = lmax;
    __syncthreads();
    for (int s = RED_THREADS / 2; s > 0; s >>= 1) {
        if (tid < s) {
            smin[tid] = (smin[tid + s] < smin[tid]) ? smin[tid + s] : smin[tid];
            smax[tid] = (smax[tid + s] > smax[tid]) ? smax[tid + s] : smax[tid];
        }
        __syncthreads();
    }
    if (tid == 0) {
        atomicMin(&ws[0], smin[0]);
        atomicMax(&ws[1], smax[0]);
    }
}

// ---------------------------------------------------------------------------
// Pass 2: 1D border-clamped linear interpolation; 4 outputs per thread.
// ---------------------------------------------------------------------------
__global__ __launch_bounds__(256)
void ds_interp_kernel(const float* __restrict__ flux,
                      const float* __restrict__ obs,
                      const u32* __restrict__ ws,
                      float* __restrict__ out) {
    const int vid = blockIdx.x * 256 + threadIdx.x;   // over (B*M)/4 = 262,144
    const int base = vid << 2;
    const int b = base >> 14;                         // M = 2^14; 4-aligned => same row
    const float* row = flux + ((size_t)b << 18);      // N = 2^18

    const float wmin = key2f(ws[0]);
    const float wmax = key2f(ws[1]);
    const float invw = 1.0f / (wmax - wmin);
    const float nm1 = (float)(N_DIM - 1);

    const float4 o = *(const float4*)(obs + base);    // global_load_b128
    float4 r;
    {
        const float* op = (const float*)&o;
        float* rp = (float*)&r;
        #pragma unroll
        for (int j = 0; j < 4; ++j) {
            float pos = (op[j] - wmin) * invw * nm1;  // matches ref up to fp assoc.
            pos = fminf(fmaxf(pos, 0.0f), nm1);
            const float i0f = floorf(pos);
            const float frac = pos - i0f;
            const int i0 = (int)i0f;
            const int i1 = min(i0 + 1, N_DIM - 1);
            const float f0 = row[i0];                 // scalar gathers, L2-resident row
            const float f1 = row[i1];
            rp[j] = f0 * (1.0f - frac) + f1 * frac;
        }
    }
    *(float4*)(out + base) = r;                       // global_store_b128
}

__global__ void ds_init_kernel(u32* __restrict__ ws) {
    ws[0] = 0xFFFFFFFFu;  // running-min identity (ordered-key space)
    ws[1] = 0x00000000u;  // running-max identity
}

extern "C" void kernel_launch(void* const* d_in, const int* in_sizes, int n_in,
                              void* d_out, int out_size, void* d_ws, size_t ws_size,
                              hipStream_t stream) {
    (void)in_sizes; (void)n_in; (void)out_size; (void)ws_size;
    const float* flux = (const float*)d_in[0];   // [B, N]
    const float* wl   = (const float*)d_in[1];   // [B, N]
    const float* obs  = (const float*)d_in[2];   // [B, M]
    float* out = (float*)d_out;                  // [B, M]
    u32* ws = (u32*)d_ws;                        // ws[0]=min key, ws[1]=max key

    ds_init_kernel<<<1, 1, 0, stream>>>(ws);
    ds_minmax_kernel<<<RED_BLOCKS, RED_THREADS, 0, stream>>>(wl, ws);
    ds_interp_kernel<<<(B_DIM * M_DIM) / 4 / 256, 256, 0, stream>>>(flux, obs, ws, out);
}